// BahdanauAttention_48850958024936
// MI455X (gfx1250) — compile-verified
//
#include <hip/hip_runtime.h>
#include <hip/hip_bf16.h>
#include <math.h>

// Problem constants (match reference)
#define BB 32
#define TT 512
#define SS 1024
#define HH 512
#define TTILE 16      // timesteps per attention block (WMMA M dim)
#define NWAVES 8      // waves per 256-thread block

typedef __attribute__((ext_vector_type(2))) float v2f;
typedef __attribute__((ext_vector_type(8))) float v8f;

// CDNA5 has a hardware transcendental V_TANH_F32; use it if the toolchain
// exposes the builtin, else fall back to libm tanhf.
__device__ __forceinline__ float fast_tanh(float x) {
#if defined(__has_builtin)
#if __has_builtin(__builtin_amdgcn_tanhf)
    return __builtin_amdgcn_tanhf(x);
#else
    return tanhf(x);
#endif
#else
    return tanhf(x);
#endif
}

// ---------------------------------------------------------------------------
// GEMM: out[m, o] = sum_h X[m, h] * W[o, h]     (torch Linear: W is [out,in])
// One wave computes one 16x16 output tile via V_WMMA_F32_16X16X4_F32.
// A (16x4) layout: lanes 0-15 -> M=lane, vgpr0=K0, vgpr1=K1; lanes 16-31 -> K2,K3.
// B (4x16) layout: lanes hold N=lane&15; vgpr0=K0(low)/K2(high), vgpr1=K1/K3.
// ---------------------------------------------------------------------------
__global__ __launch_bounds__(256) void gemm_nt_wmma_f32(
    const float* __restrict__ X, const float* __restrict__ W,
    float* __restrict__ out, int M)
{
    const int lane = threadIdx.x & 31;
    const int wid  = threadIdx.x >> 5;
    const int ntiles_o = HH / 16;                    // 32 column tiles
    const int tile = blockIdx.x * NWAVES + wid;
    const int m0 = (tile / ntiles_o) * 16;
    const int o0 = (tile % ntiles_o) * 16;
    if (m0 >= M) return;                             // wave-uniform

    const int idx = lane & 15;
    const int kk  = (lane < 16) ? 0 : 2;
    const float* xr = X + (size_t)(m0 + idx) * HH;   // A row for this lane
    const float* wr = W + (size_t)(o0 + idx) * HH;   // B col (= W row) for this lane

    v8f c = {};
    #pragma unroll 8
    for (int k0 = 0; k0 < HH; k0 += 4) {
        v2f a, b;
        a[0] = xr[k0 + kk];     a[1] = xr[k0 + kk + 1];
        b[0] = wr[k0 + kk];     b[1] = wr[k0 + kk + 1];
        c = __builtin_amdgcn_wmma_f32_16x16x4_f32(false, a, false, b,
                                                  (short)0, c, false, false);
    }

    const int roff = (lane < 16) ? 0 : 8;            // C/D: vgpr r -> M=r or r+8
    #pragma unroll
    for (int r = 0; r < 8; ++r)
        out[(size_t)(m0 + r + roff) * HH + o0 + idx] = c[r];
}

// ---------------------------------------------------------------------------
// Fused attention for one (batch b, 16-timestep tile):
//   phase 1: scores[t][s] = sum_h v[h]*tanh(Whe[b,s,h] + Wsq[b,t0+t,h])
//   phase 2: masked softmax over s per t row
//   phase 3: ctx[16,512] = attn[16,1024] @ Eo[b]  via WMMA f32 16x16x4
// ---------------------------------------------------------------------------
__global__ __launch_bounds__(256) void attn_fused_wmma(
    const float* __restrict__ Whe, const float* __restrict__ Wsq,
    const float* __restrict__ Eo,  const float* __restrict__ v,
    const int* __restrict__ src_len,
    float* __restrict__ ctx_out, float* __restrict__ attn_out)
{
    __shared__ float wsq_s[TTILE][HH + 1];   // 32.8 KB
    __shared__ float v_s[HH];                //  2.0 KB
    __shared__ float sc[TTILE][SS + 1];      // 65.6 KB (pad -> conflict-free A reads)

    const int tid  = threadIdx.x;
    const int lane = tid & 31;
    const int wid  = tid >> 5;
    const int b  = blockIdx.x / (TT / TTILE);
    const int t0 = (blockIdx.x % (TT / TTILE)) * TTILE;
    const int len = src_len[b];

    for (int i = tid; i < TTILE * HH; i += 256) {
        int t = i / HH, h = i % HH;
        wsq_s[t][h] = Wsq[((size_t)b * TT + t0 + t) * HH + h];
    }
    for (int i = tid; i < HH; i += 256) v_s[i] = v[i];
    __syncthreads();

    // ---- phase 1: scores (lanes strided over h, coalesced Whe reads) ----
    for (int s = wid; s < SS; s += NWAVES) {
        const float* wr = Whe + ((size_t)b * SS + s) * HH;
        float wh[16];
        #pragma unroll
        for (int k = 0; k < 16; ++k) wh[k] = wr[lane + 32 * k];
        float acc[TTILE];
        #pragma unroll
        for (int t = 0; t < TTILE; ++t) acc[t] = 0.0f;
        #pragma unroll
        for (int k = 0; k < 16; ++k) {
            const float w  = wh[k];
            const float vv = v_s[lane + 32 * k];
            #pragma unroll
            for (int t = 0; t < TTILE; ++t)
                acc[t] = fmaf(vv, fast_tanh(w + wsq_s[t][lane + 32 * k]), acc[t]);
        }
        #pragma unroll
        for (int t = 0; t < TTILE; ++t) {
            float r = acc[t];
            #pragma unroll
            for (int m = 16; m >= 1; m >>= 1) r += __shfl_xor(r, m, 32);
            if (lane == 0) sc[t][s] = r;
        }
    }
    __syncthreads();

    // ---- phase 2: masked softmax per t row (2 rows per wave) ----
    for (int t = wid; t < TTILE; t += NWAVES) {
        float mx = -INFINITY;
        for (int s = lane; s < SS; s += 32) {
            float x = (s < len) ? sc[t][s] : -INFINITY;
            mx = fmaxf(mx, x);
        }
        #pragma unroll
        for (int m = 16; m >= 1; m >>= 1) mx = fmaxf(mx, __shfl_xor(mx, m, 32));
        float sum = 0.0f;
        for (int s = lane; s < SS; s += 32) {
            float e = (s < len) ? __expf(sc[t][s] - mx) : 0.0f;
            sc[t][s] = e;
            sum += e;
        }
        #pragma unroll
        for (int m = 16; m >= 1; m >>= 1) sum += __shfl_xor(sum, m, 32);
        const float inv = 1.0f / sum;
        for (int s = lane; s < SS; s += 32) sc[t][s] *= inv;
    }
    __syncthreads();

    // second output: attention weights of the LAST timestep (t = TT-1)
    if (t0 == TT - TTILE) {
        for (int s = tid; s < SS; s += 256)
            attn_out[(size_t)b * SS + s] = sc[TTILE - 1][s];
    }

    // ---- phase 3: ctx = attn @ Eo, WMMA f32 16x16x4; 4 col-tiles per wave ----
    const float* eb = Eo + (size_t)b * SS * HH;
    const int n    = lane & 15;
    const int kk   = (lane < 16) ? 0 : 2;
    const int roff = (lane < 16) ? 0 : 8;

    v8f c[4] = {};
    for (int k0 = 0; k0 < SS; k0 += 4) {
        v2f a;                                        // A = attn tile from LDS
        a[0] = sc[n][k0 + kk];
        a[1] = sc[n][k0 + kk + 1];
        #pragma unroll
        for (int j = 0; j < 4; ++j) {
            const int h0 = (wid * 4 + j) * 16;
            v2f bv;                                   // B = Eo[s, h] (row-major)
            bv[0] = eb[(size_t)(k0 + kk)     * HH + h0 + n];
            bv[1] = eb[(size_t)(k0 + kk + 1) * HH + h0 + n];
            c[j] = __builtin_amdgcn_wmma_f32_16x16x4_f32(false, a, false, bv,
                                                         (short)0, c[j], false, false);
        }
    }

    #pragma unroll
    for (int j = 0; j < 4; ++j) {
        const int h0 = (wid * 4 + j) * 16;
        #pragma unroll
        for (int r = 0; r < 8; ++r) {
            const int t = r + roff;
            ctx_out[((size_t)b * TT + t0 + t) * HH + h0 + n] = c[j][r];
        }
    }
}

// ---------------------------------------------------------------------------
extern "C" void kernel_launch(void* const* d_in, const int* in_sizes, int n_in,
                              void* d_out, int out_size, void* d_ws, size_t ws_size,
                              hipStream_t stream)
{
    const float* query   = (const float*)d_in[0];  // [B,T,H]
    const float* eo      = (const float*)d_in[1];  // [B,S,H]
    const int*   src_len = (const int*)  d_in[2];  // [B]
    const float* Ws      = (const float*)d_in[3];  // [H,H]
    const float* Wh      = (const float*)d_in[4];  // [H,H]
    const float* v       = (const float*)d_in[5];  // [H]

    float* ctx  = (float*)d_out;                       // [B,T,H]
    float* attn = ctx + (size_t)BB * TT * HH;          // [B,S]

    float* Whe = (float*)d_ws;                         // [B,S,H]  64 MB
    float* Wsq = Whe + (size_t)BB * SS * HH;           // [B,T,H]  32 MB

    // Whe = Eo @ Wh^T
    {
        const int M = BB * SS;
        const int tiles = (M / 16) * (HH / 16);
        gemm_nt_wmma_f32<<<tiles / NWAVES, 256, 0, stream>>>(eo, Wh, Whe, M);
    }
    // Wsq = Q @ Ws^T
    {
        const int M = BB * TT;
        const int tiles = (M / 16) * (HH / 16);
        gemm_nt_wmma_f32<<<tiles / NWAVES, 256, 0, stream>>>(query, Ws, Wsq, M);
    }
    // Fused scores + softmax + context
    attn_fused_wmma<<<BB * (TT / TTILE), 256, 0, stream>>>(
        Whe, Wsq, eo, v, src_len, ctx, attn);
}